// ShortConv_32126355374151
// MI455X (gfx1250) — compile-verified
//
#include <hip/hip_runtime.h>

typedef float v4f __attribute__((ext_vector_type(4)));

#define TT      16            // output timesteps per block
#define ROWS    (TT + 3)      // input rows needed (K-1 = 3 halo)
#define CH      512           // channels per block
#define THREADS 128           // CH / 4 channels-per-thread

__global__ __launch_bounds__(THREADS)
void ShortConv_depthwise_causal_kernel(const float* __restrict__ x,
                                       const float* __restrict__ w,
                                       float* __restrict__ y,
                                       int T, int C)
{
    // 19 rows x 512 channels x 4B = 38 KB of LDS staging
    __shared__ float tile[ROWS * CH];

    const int tid = threadIdx.x;
    const int t0  = blockIdx.x * TT;                 // time tile start
    const int c0  = blockIdx.y * CH + tid * 4;       // this thread's 4 channels
    const int b   = blockIdx.z;

    // LDS byte offset of our staging buffer: generic AS pointer to LDS has
    // the LDS byte address in its low 32 bits (ISA 10.2 aperture mapping),
    // matching the async load, which takes its LDS byte address from the VDST VGPR.
    const unsigned lds_base = (unsigned)(unsigned long long)(&tile[0]);
    const unsigned col_b    = (unsigned)(tid * 16);

    const unsigned long long xrow0 =
        ((unsigned long long)b * (unsigned)T) * (unsigned)C + (unsigned)c0;

    // ---- Stage ROWS rows of x into LDS via CDNA5 async global->LDS loads ----
#pragma unroll
    for (int r = 0; r < ROWS; ++r) {
        const int t = t0 - 3 + r;
        const unsigned lds_addr = lds_base + (unsigned)(r * (CH * 4)) + col_b;
        if (t >= 0) {
            const float* gp = x + xrow0 + (unsigned long long)t * (unsigned)C;
            // vdst = LDS byte address (VGPR), vaddr = 64-bit global address (VGPR pair)
            asm volatile("global_load_async_to_lds_b128 %0, %1, off"
                         :: "v"(lds_addr), "v"(gp)
                         : "memory");
        } else {
            // causal left padding: zero-fill (plain DS store; same-wave LDS ops
            // are in-order, so later LDS reads see this without extra waits)
            *(v4f*)(&tile[r * CH + tid * 4]) = (v4f)0.0f;
        }
    }

    // Weights for this thread's 4 channels (tiny; L2-resident)
    const v4f w0 = *(const v4f*)(w + 0ull * C + c0);
    const v4f w1 = *(const v4f*)(w + 1ull * C + c0);
    const v4f w2 = *(const v4f*)(w + 2ull * C + c0);
    const v4f w3 = *(const v4f*)(w + 3ull * C + c0);

    // Wait for our wave's async LDS writes to land (compiler can't track the
    // asm-issued async ops; no cross-thread LDS sharing, so no barrier needed).
    asm volatile("s_wait_asynccnt 0x0" ::: "memory");

    // ---- Rolling 4-register window over LDS rows ----
    v4f a0 = *(const v4f*)(&tile[0 * CH + tid * 4]);
    v4f a1 = *(const v4f*)(&tile[1 * CH + tid * 4]);
    v4f a2 = *(const v4f*)(&tile[2 * CH + tid * 4]);
    v4f a3 = *(const v4f*)(&tile[3 * CH + tid * 4]);

    float* yb = y + ((unsigned long long)b * (unsigned)T + (unsigned)t0) * (unsigned)C
                  + (unsigned)c0;

#pragma unroll
    for (int j = 0; j < TT; ++j) {
        v4f acc = a0 * w0 + a1 * w1 + a2 * w2 + a3 * w3;   // fused into v_fma chains
        // y is write-once / never re-read: non-temporal store keeps the 192 MB
        // L2 free to hold x (134 MB), so halo-row re-reads stay L2-resident.
        __builtin_nontemporal_store(acc, (v4f*)(yb + (unsigned long long)j * (unsigned)C));
        a0 = a1; a1 = a2; a2 = a3;
        if (j + 1 < TT)
            a3 = *(const v4f*)(&tile[(j + 4) * CH + tid * 4]);
    }
}

extern "C" void kernel_launch(void* const* d_in, const int* in_sizes, int n_in,
                              void* d_out, int out_size, void* d_ws, size_t ws_size,
                              hipStream_t stream) {
    const float* x = (const float*)d_in[0];   // (B, T, C) fp32
    const float* w = (const float*)d_in[1];   // (K=4, 1, C) fp32
    float*       y = (float*)d_out;           // (B, T, C) fp32

    const int K = 4;
    const int C = in_sizes[1] / K;            // 2048
    const int B = 4;
    const int T = in_sizes[0] / (B * C);      // 4096

    dim3 grid(T / TT, C / CH, B);             // (256, 4, 4)
    ShortConv_depthwise_causal_kernel<<<grid, THREADS, 0, stream>>>(x, w, y, T, C);
}